// CDDPE_82394652606946
// MI455X (gfx1250) — compile-verified
//
#include <hip/hip_runtime.h>
#include <hip/hip_bf16.h>
#include <math.h>

typedef __attribute__((ext_vector_type(2))) float v2f;
typedef __attribute__((ext_vector_type(8))) float v8f;
typedef __attribute__((ext_vector_type(2))) float f32x2;

#define IMG 256
#define IC 3
#define HID 64
#define NB 16
#define NE 4
#define D_FLAT (IC*IMG*IMG)   // 196608

// ---------------- Kernel 1: per-(b,c) plane sums of ref -> s = sum/256 ----------------
__global__ __launch_bounds__(256) void k_ref_sums(const float* __restrict__ ref, float* __restrict__ s) {
  __shared__ float red[256];
  const int plane = blockIdx.x;                 // 0..47  (b*3+c)
  const float* p = ref + (size_t)plane * (IMG*IMG);
  float a = 0.f;
  for (int i = threadIdx.x; i < IMG*IMG; i += 256) a += p[i];
  red[threadIdx.x] = a;
  __syncthreads();
  for (int st = 128; st > 0; st >>= 1) {
    if (threadIdx.x < st) red[threadIdx.x] += red[threadIdx.x + st];
    __syncthreads();
  }
  if (threadIdx.x == 0) s[plane] = red[0] * (1.0f / 256.0f);   // DC bin of ortho rfft2
}

// ---------------- Kernel 2: DC-bin MLP chain -> prompt[16][3]; zero logit accum ----------------
__global__ __launch_bounds__(64) void k_prompt(
    const float* __restrict__ s,
    const float* __restrict__ w1, const float* __restrict__ b1,
    const float* __restrict__ w2, const float* __restrict__ b2,
    const float* __restrict__ ew, const float* __restrict__ eb,
    const float* __restrict__ fre,
    float* __restrict__ prompt, float* __restrict__ logit_acc) {
  const int t = threadIdx.x;
  for (int i = t; i < 128; i += 64) logit_acc[i] = 0.f;   // zero 16*8 accumulators
  if (t < NB) {
    float p12[12];
    p12[0] = p12[6] = s[t*3 + 0];
    p12[1] = p12[7] = s[t*3 + 1];
    p12[2] = p12[8] = s[t*3 + 2];
    p12[3] = p12[4] = p12[5] = p12[9] = p12[10] = p12[11] = 0.f;
    float m1[6];
    for (int o = 0; o < 6; ++o) {
      float a = b1[o];
      for (int i = 0; i < 12; ++i) a = fmaf(w1[o*12 + i], p12[i], a);
      m1[o] = fmaxf(a, 0.f);
    }
    float m2[6];
    for (int o = 0; o < 6; ++o) {
      float a = b2[o];
      for (int i = 0; i < 6; ++i) a = fmaf(w2[o*6 + i], m1[i], a);
      m2[o] = a;
    }
    float e1[6];
    for (int i = 0; i < 6; ++i) e1[i] = m2[i] * fre[(size_t)i * (IMG * (IMG/2 + 1))]; // fre_prompt[c][0][0]
    float e2[6];
    for (int o = 0; o < 6; ++o) {
      float a = eb[o];
      for (int i = 0; i < 6; ++i) a = fmaf(ew[o*6 + i], e1[i], a);
      e2[o] = fmaxf(a, 0.f);
    }
    // spatial mean == Re(DC)/sqrt(H*W); softmax over channels
    float mv0 = e2[0] * (1.f/256.f), mv1 = e2[1] * (1.f/256.f), mv2 = e2[2] * (1.f/256.f);
    float mx = fmaxf(mv0, fmaxf(mv1, mv2));
    float x0 = expf(mv0 - mx), x1 = expf(mv1 - mx), x2 = expf(mv2 - mx);
    float den = x0 + x1 + x2;
    prompt[t*3 + 0] = x0 / den;
    prompt[t*3 + 1] = x1 / den;
    prompt[t*3 + 2] = x2 / den;
  }
}

// ---------------- Kernel 3: gating dot products: clean/noise logits ----------------
__global__ __launch_bounds__(256) void k_gating(
    const float* __restrict__ x, const float* __restrict__ wg, const float* __restrict__ wn,
    const float* __restrict__ prompt, float* __restrict__ acc) {
  __shared__ float red[256 * 8];
  const int b = blockIdx.y;
  const int base = blockIdx.x * 4096;
  const float pr0 = 1.f + prompt[b*3 + 0];
  const float pr1 = 1.f + prompt[b*3 + 1];
  const float pr2 = 1.f + prompt[b*3 + 2];
  const float* xb = x + (size_t)b * D_FLAT;
  float p[8] = {0.f,0.f,0.f,0.f,0.f,0.f,0.f,0.f};
  for (int i = threadIdx.x; i < 4096; i += 256) {
    int d = base + i;
    int c = d >> 16;                                  // d / 65536
    float scale = (c == 0) ? pr0 : ((c == 1) ? pr1 : pr2);
    float xds = xb[d] * scale;
    const float* g = wg + (size_t)d * 4;
    const float* n = wn + (size_t)d * 4;
    p[0] = fmaf(xds, g[0], p[0]); p[1] = fmaf(xds, g[1], p[1]);
    p[2] = fmaf(xds, g[2], p[2]); p[3] = fmaf(xds, g[3], p[3]);
    p[4] = fmaf(xds, n[0], p[4]); p[5] = fmaf(xds, n[1], p[5]);
    p[6] = fmaf(xds, n[2], p[6]); p[7] = fmaf(xds, n[3], p[7]);
  }
  for (int j = 0; j < 8; ++j) red[threadIdx.x*8 + j] = p[j];
  __syncthreads();
  for (int st = 128; st > 0; st >>= 1) {
    if (threadIdx.x < st)
      for (int j = 0; j < 8; ++j) red[threadIdx.x*8 + j] += red[(threadIdx.x + st)*8 + j];
    __syncthreads();
  }
  if (threadIdx.x < 8) atomicAdd(&acc[b*8 + threadIdx.x], red[threadIdx.x]);
}

// ---------------- Kernel 4: noisy top-k gating + load-balance loss ----------------
__device__ __forceinline__ float cv_squared4(const float* v) {
  float mean = (v[0] + v[1] + v[2] + v[3]) * 0.25f;
  float var = 0.f;
  for (int i = 0; i < 4; ++i) { float d = v[i] - mean; var += d * d; }
  var *= (1.f / 3.f);                                 // ddof=1
  return var / (mean * mean + 1e-10f);
}

__global__ __launch_bounds__(32) void k_moe_gates(
    const float* __restrict__ noise, const float* __restrict__ lac,
    float* __restrict__ gates, float* __restrict__ loss_out) {
  __shared__ float gmat[NB][NE];
  __shared__ float pmat[NB][NE];
  const int t = threadIdx.x;
  if (t < NB) {
    float cl[NE], st[NE], nl[NE];
    for (int e = 0; e < NE; ++e) {
      cl[e] = lac[t*8 + e];
      float raw = lac[t*8 + 4 + e];
      float sp = fmaxf(raw, 0.f) + log1pf(expf(-fabsf(raw)));   // stable softplus
      st[e] = sp + 0.01f;
      nl[e] = fmaf(noise[t*NE + e], st[e], cl[e]);
    }
    float v[NE] = {nl[0], nl[1], nl[2], nl[3]};
    int idx[NE] = {0, 1, 2, 3};
    for (int i = 0; i < 3; ++i) {                     // selection sort, top-3
      int m = i;
      for (int j = i + 1; j < NE; ++j) if (v[j] > v[m]) m = j;
      float tv = v[i]; v[i] = v[m]; v[m] = tv;
      int ti = idx[i]; idx[i] = idx[m]; idx[m] = ti;
    }
    float e0 = 1.f, e1 = expf(v[1] - v[0]);
    float den = e0 + e1;
    for (int e = 0; e < NE; ++e) gmat[t][e] = 0.f;
    gmat[t][idx[0]] = e0 / den;
    gmat[t][idx[1]] = e1 / den;
    const float thr_in = v[2], thr_out = v[1];
    for (int e = 0; e < NE; ++e) {
      bool is_in = nl[e] > thr_in;
      float z = (cl[e] - (is_in ? thr_in : thr_out)) / st[e];
      pmat[t][e] = 0.5f * erfcf(-z * 0.70710678118654752440f);  // norm.cdf
      gates[t*NE + e] = gmat[t][e];
    }
  }
  __syncthreads();
  if (t == 0) {
    float imp[NE] = {0,0,0,0}, ld[NE] = {0,0,0,0};
    for (int b = 0; b < NB; ++b)
      for (int e = 0; e < NE; ++e) { imp[e] += gmat[b][e]; ld[e] += pmat[b][e]; }
    loss_out[0] = (cv_squared4(imp) + cv_squared4(ld)) * 0.01f;
  }
}

// ---------------- Kernel 5: fused top-2 expert convs (WMMA conv1 + packed-VALU conv2) ----------------
#define TW 16
#define TH 8
#define HR_H (TH + 2)          // 10
#define HR_W (TW + 2)          // 18
#define HP   (HR_H * HR_W)     // 180 live h-pixels (with halo)
#define HROWS 192              // 12 full M-tiles of 16; rows 180..191 write-only scratch
#define XR_H (TH + 4)          // 12
#define XR_W (TW + 4)          // 20
#define HPAD 68                // 272 B pixel stride: 16B-aligned rows, conflict-free

// compile-time im2col offset for k = c*9 + dy*3 + dx
#define OFFK(k) (((k)/9) * (XR_H*XR_W) + ((((k)%9))/3) * XR_W + (((k)%9)%3))

__global__ __launch_bounds__(128) void k_experts(
    const float* __restrict__ x, const float* __restrict__ ew1, const float* __restrict__ eb1,
    const float* __restrict__ ew2, const float* __restrict__ eb2,
    const float* __restrict__ gates, float* __restrict__ y) {
  __shared__ __align__(16) float hbuf[HROWS * HPAD]; // 52.2 KB hidden tile (12 full M-tiles)
  __shared__ float Bmt[HID * 28];                    //  7.2 KB conv1 weights [o][k], col 27 zeroed
  __shared__ __align__(16) float w2t[9 * HID];       //  2.3 KB conv2 weights, [tap][ch]
  __shared__ float ebs[HID];                         //  256 B conv1 bias
  // x patch aliases the write-only tail of hbuf (rows 180..191 = 816 floats >= 720 needed).
  // All xs reads finish before the first expert's staging barrier; conv1 stores come after it.
  float* xs = &hbuf[HP * HPAD];

  const int tid = threadIdx.x;
  const int b   = blockIdx.z;
  const int tx0 = blockIdx.x * TW;
  const int ty0 = blockIdx.y * TH;
  const bool border = (blockIdx.x == 0) | (blockIdx.x == IMG/TW - 1) |
                      (blockIdx.y == 0) | (blockIdx.y == IMG/TH - 1);

  // Stage x patch (zero 'SAME' padding at image edges)
  const float* xb = x + (size_t)b * D_FLAT;
  for (int i = tid; i < IC * XR_H * XR_W; i += 128) {
    int c = i / (XR_H * XR_W);
    int r = i % (XR_H * XR_W);
    int gy = ty0 - 2 + r / XR_W;
    int gx = tx0 - 2 + r % XR_W;
    float v = 0.f;
    if (gy >= 0 && gy < IMG && gx >= 0 && gx < IMG) v = xb[c * IMG*IMG + gy * IMG + gx];
    xs[i] = v;
  }
  __syncthreads();

  const int wv    = tid >> 5;
  const int lane  = tid & 31;
  const int mrow  = lane & 15;
  const bool lo16 = (lane < 16);
  const int hi8   = lo16 ? 0 : 8;          // D-matrix upper-half M offset
  const int khalf = lo16 ? 0 : 2;          // A/B K-offset for upper half-wave (ISA §7.12.2)

  // Hoist A fragments (im2col of x) into registers once per block: expert-invariant.
  // Each wave owns M-tiles {wv, wv+4, wv+8}; im2col offsets are compile-time constants
  // selected per half-wave with one cndmask (no runtime div/mod).
  v2f av[3][7];
  #pragma unroll
  for (int tt = 0; tt < 3; ++tt) {
    const int p0 = (wv + tt*4) * 16;
    int p = p0 + mrow; if (p > HP - 1) p = HP - 1;   // clamp: rows >=180 get duplicate (valid) A rows
    const int pb = (p / HR_W) * XR_W + (p % HR_W);
    #pragma unroll
    for (int ks = 0; ks < 7; ++ks) {
      const int kx0 = lo16 ? OFFK(ks*4 + 0) : OFFK(ks*4 + 2);
      const int kx1 = lo16 ? OFFK(ks*4 + 1) : OFFK((ks*4 + 3 > 26) ? 26 : ks*4 + 3); // k=27 nullified by zero B col
      av[tt][ks].x = xs[pb + kx0];
      av[tt][ks].y = xs[pb + kx1];
    }
  }

  float yacc = 0.f;

  for (int e = 0; e < NE; ++e) {
    const float gate = gates[b*NE + e];
    if (gate == 0.f) continue;             // top-2 sparsity: block-uniform branch
    __builtin_prefetch(ew1 + (size_t)((e + 1) & 3) * HID * 27, 0, 1);

    // Stage per-expert weights: Bmt[o][k] transposed, padded to 28 (col 27 = 0); bias to LDS
    for (int i = tid; i < HID * 28; i += 128) {
      int o = i / 28, k = i % 28;
      Bmt[i] = (k < 27) ? ew1[(size_t)(e * HID + o) * 27 + k] : 0.f;
    }
    for (int i = tid; i < 9 * HID; i += 128) {
      int tap = i >> 6, c = i & 63;
      w2t[i] = ew2[e * HID * 9 + c * 9 + tap];                        // transpose to [tap][ch]
    }
    if (tid < HID) ebs[tid] = eb1[e * HID + tid];
    __syncthreads();

    // -------- conv1: implicit GEMM, M=16 h-pixels, N=64, K=27 (pad 28) via V_WMMA_F32_16X16X4_F32
    #pragma unroll
    for (int nt = 0; nt < 4; ++nt) {
      const int n = nt * 16 + mrow;
      const float bias = ebs[n];
      v2f bf7[7];
      #pragma unroll
      for (int ks = 0; ks < 7; ++ks)                  // contiguous (k,k+1) pair: one ds_load_b64
        bf7[ks] = *(const v2f*)&Bmt[n * 28 + ks*4 + khalf];
      #pragma unroll
      for (int tt = 0; tt < 3; ++tt) {
        v8f acc = {};
        #pragma unroll
        for (int ks = 0; ks < 7; ++ks)
          acc = __builtin_amdgcn_wmma_f32_16x16x4_f32(
              false, av[tt][ks], false, bf7[ks], (short)0, acc, false, false);
        const int base = ((wv + tt*4) * 16 + hi8) * HPAD + n;
        #pragma unroll
        for (int r = 0; r < 8; ++r)                   // unconditional: rows >=180 are scratch
          hbuf[base + r * HPAD] = fmaxf(acc[r] + bias, 0.f);
      }
    }
    __syncthreads();

    // Zero h where (gy,gx) falls outside the image: conv2 'SAME' pads h with zeros.
    if (border) {
      const float4 z4 = {0.f, 0.f, 0.f, 0.f};
      for (int i = tid; i < HP; i += 128) {
        int gy = ty0 - 1 + i / HR_W;
        int gx = tx0 - 1 + i % HR_W;
        if ((unsigned)gy >= (unsigned)IMG || (unsigned)gx >= (unsigned)IMG) {
          #pragma unroll
          for (int c = 0; c < HID; c += 4) *(float4*)&hbuf[i * HPAD + c] = z4;
        }
      }
    }
    __syncthreads();

    // -------- conv2: N=1 GEMM is WMMA-hostile -> packed VALU (v_pk_fma_f32) over LDS h tile
    {
      const int oy = tid >> 4, ox = tid & 15;
      f32x2 s0 = {0.f, 0.f}, s1 = {0.f, 0.f}, s2 = {0.f, 0.f}, s3 = {0.f, 0.f};
      #pragma unroll
      for (int ky = 0; ky < 3; ++ky)
        #pragma unroll
        for (int kx = 0; kx < 3; ++kx) {
          const f32x2* h2 = (const f32x2*)&hbuf[((oy + ky) * HR_W + ox + kx) * HPAD];
          const f32x2* w2 = (const f32x2*)&w2t[(ky * 3 + kx) * HID];
          #pragma unroll
          for (int c = 0; c < HID/2; c += 4) {        // 4 independent packed accumulators
            s0 = __builtin_elementwise_fma(h2[c + 0], w2[c + 0], s0);
            s1 = __builtin_elementwise_fma(h2[c + 1], w2[c + 1], s1);
            s2 = __builtin_elementwise_fma(h2[c + 2], w2[c + 2], s2);
            s3 = __builtin_elementwise_fma(h2[c + 3], w2[c + 3], s3);
          }
        }
      f32x2 sv = (s0 + s1) + (s2 + s3);
      float acc2 = eb2[e] + sv.x + sv.y;
      yacc = fmaf(gate, acc2, yacc);
    }
    __syncthreads();
  }

  const int oy = tid >> 4, ox = tid & 15;
  y[(size_t)b * IMG*IMG + (ty0 + oy) * IMG + (tx0 + ox)] = yacc;
}

// ---------------- Launch ----------------
extern "C" void kernel_launch(void* const* d_in, const int* in_sizes, int n_in,
                              void* d_out, int out_size, void* d_ws, size_t ws_size,
                              hipStream_t stream) {
  const float* x       = (const float*)d_in[0];
  const float* ref     = (const float*)d_in[1];
  const float* noise   = (const float*)d_in[2];
  const float* mod_w1  = (const float*)d_in[3];
  const float* mod_b1  = (const float*)d_in[4];
  const float* mod_w2  = (const float*)d_in[5];
  const float* mod_b2  = (const float*)d_in[6];
  const float* enh_w   = (const float*)d_in[7];
  const float* enh_b   = (const float*)d_in[8];
  const float* fre     = (const float*)d_in[9];
  const float* w_gate  = (const float*)d_in[10];
  const float* w_noise = (const float*)d_in[11];
  const float* ew1     = (const float*)d_in[12];
  const float* eb1     = (const float*)d_in[13];
  const float* ew2     = (const float*)d_in[14];
  const float* eb2     = (const float*)d_in[15];
  float* out = (float*)d_out;
  float* ws  = (float*)d_ws;
  // ws layout (floats): [0..47] plane sums, [64..111] prompt, [128..255] logit accum, [256..319] gates
  float* ws_s      = ws;
  float* ws_prompt = ws + 64;
  float* ws_logit  = ws + 128;
  float* ws_gates  = ws + 256;

  k_ref_sums<<<48, 256, 0, stream>>>(ref, ws_s);
  k_prompt<<<1, 64, 0, stream>>>(ws_s, mod_w1, mod_b1, mod_w2, mod_b2,
                                 enh_w, enh_b, fre, ws_prompt, ws_logit);
  k_gating<<<dim3(48, NB), 256, 0, stream>>>(x, w_gate, w_noise, ws_prompt, ws_logit);
  k_moe_gates<<<1, 32, 0, stream>>>(noise, ws_logit, ws_gates, out + (size_t)NB * IMG * IMG);
  k_experts<<<dim3(IMG / TW, IMG / TH, NB), 128, 0, stream>>>(
      x, ew1, eb1, ew2, eb2, ws_gates, out);
}